// ConvLayer_4140348473931
// MI455X (gfx1250) — compile-verified
//
#include <hip/hip_runtime.h>

typedef __attribute__((ext_vector_type(2))) float v2f;
typedef __attribute__((ext_vector_type(8))) float v8f;
typedef __attribute__((ext_vector_type(4))) int   v4i;
typedef __attribute__((address_space(1))) v4i    *as1_v4i_p;  // global int4*
typedef __attribute__((address_space(3))) v4i    *as3_v4i_p;  // LDS int4*

#define IH 4096
#define IW 8192
#define OH 4094
#define OW 8190

#define TY 16        // output rows per workgroup
#define TX 128       // output cols per workgroup (8 waves x 16)
#define LROWS 18     // staged input rows (TY + 2)
#define LCOLS 132    // staged input cols: 16B-aligned rows; stride 132 % 64 == 4
                     // -> conflict-free b64 fragment reads
#define KPAD 20      // horizontal-window K (18) padded to 5 chunks of 4

__global__ __launch_bounds__(256)
void conv3x3_wmma_kernel(const float* __restrict__ X,
                         const float* __restrict__ W,
                         const float* __restrict__ Bv,
                         float* __restrict__ out)
{
    __shared__ float tile[LROWS][LCOLS];     // input stage    (9504 B)
    __shared__ float btab[3][16][KPAD];      // Toeplitz B     (3840 B)  [i][n][k]
    __shared__ float cs[TY][LCOLS];          // output stage   (8448 B)

    const int tid = threadIdx.x;
    const int x0  = blockIdx.x * TX;
    const int y0  = blockIdx.y * TY;

    // ---------- stage 18 x 132 input tile: global -> LDS (ASYNCcnt path) ----------
    for (int idx = tid; idx < LROWS * (LCOLS / 4); idx += 256) {
        int r  = idx / (LCOLS / 4);
        int c4 = idx % (LCOLS / 4);
        int gr = y0 + r;      if (gr > IH - 1) gr = IH - 1;
        int gc = x0 + c4 * 4; if (gc > IW - 4) gc = IW - 4;
        const float* gp = X + (size_t)gr * IW + gc;
        float* lp = &tile[r][c4 * 4];
#if __has_builtin(__builtin_amdgcn_global_load_async_to_lds_b128)
        __builtin_amdgcn_global_load_async_to_lds_b128(
            (as1_v4i_p)gp, (as3_v4i_p)lp, 0, 0);
#else
        *(float4*)lp = *(const float4*)gp;
#endif
    }

    // ---------- build banded-Toeplitz B tables once (3 x 16 x 20) ----------
    // btab[i][n][k] = W[i][k-n] for k-n in [0,2], else 0
    for (int idx = tid; idx < 3 * 16 * KPAD; idx += 256) {
        int i   = idx / (16 * KPAD);
        int rem = idx - i * (16 * KPAD);
        int n   = rem / KPAD;
        int k   = rem - n * KPAD;
        int d   = k - n;
        float v = 0.0f;
        if (d == 0) v = W[i * 3 + 0];
        else if (d == 1) v = W[i * 3 + 1];
        else if (d == 2) v = W[i * 3 + 2];
        btab[i][n][k] = v;
    }

#if __has_builtin(__builtin_amdgcn_s_wait_asynccnt)
    __builtin_amdgcn_s_wait_asynccnt(0);
#else
    asm volatile("s_wait_asynccnt 0" ::: "memory");
#endif
    __syncthreads();

    // ---------- per-wave 16x16 output tile via 15x V_WMMA_F32_16X16X4_F32 ----------
    const int lane = tid & 31;
    const int wv   = tid >> 5;       // wave id 0..7
    const int hi   = lane >> 4;      // lane half: K offset (A/B) / row half (C)
    const int m    = lane & 15;      // A row index == B/C column index n
    const int xl   = wv * 16;        // this wave's column offset inside the LDS tile

    const float bias = Bv[0];

    v8f c = {0.f, 0.f, 0.f, 0.f, 0.f, 0.f, 0.f, 0.f};

#pragma unroll
    for (int i = 0; i < 3; ++i) {
#pragma unroll
        for (int kc = 0; kc < 5; ++kc) {
            const int k0 = kc * 4 + 2 * hi;   // this lane's first K index of the chunk
            // A: lane holds X[y0+i+m][x0 + xl + k0 .. +1]   (conflict-free ds_load_b64)
            v2f a = *(const v2f*)&tile[i + m][xl + k0];
            // B: lane holds T_i[k0..k0+1][m]                (conflict-free ds_load_b64)
            v2f b = *(const v2f*)&btab[i][m][k0];
            c = __builtin_amdgcn_wmma_f32_16x16x4_f32(
                    /*neg_a=*/false, a, /*neg_b=*/false, b,
                    /*c_mod=*/(short)0, c, /*reuse_a=*/false, /*reuse_b=*/false);
        }
    }

    // ---------- stage C (+bias) to LDS, then coalesced b64 stores ----------
    // C layout: VGPR v -> row v + 8*hi, lane -> col m
#pragma unroll
    for (int v = 0; v < 8; ++v)
        cs[v + 8 * hi][xl + m] = c[v] + bias;

    __syncthreads();

    // 16 rows x 64 float2 chunks = 1024; 4 rounds of 256 threads.
    // Every out-row base (oy*8190*4 bytes) and ox (even) are 8B-aligned -> b64 ok.
    for (int idx = tid; idx < TY * (TX / 2); idx += 256) {
        int r  = idx >> 6;          // 0..15
        int c2 = idx & 63;          // 0..63
        int oy = y0 + r;
        int ox = x0 + c2 * 2;
        v2f val = *(const v2f*)&cs[r][c2 * 2];
        if (oy < OH) {
            if (ox + 1 < OW)       *(v2f*)(out + (size_t)oy * OW + ox) = val;
            else if (ox < OW)      out[(size_t)oy * OW + ox] = val.x;
        }
    }
}

extern "C" void kernel_launch(void* const* d_in, const int* in_sizes, int n_in,
                              void* d_out, int out_size, void* d_ws, size_t ws_size,
                              hipStream_t stream) {
    (void)in_sizes; (void)n_in; (void)d_ws; (void)ws_size; (void)out_size;
    const float* X  = (const float*)d_in[0];   // [4096, 8192] f32
    const float* W  = (const float*)d_in[1];   // [3, 3] f32
    const float* Bv = (const float*)d_in[2];   // [1] f32
    float* out = (float*)d_out;                // [4094, 8190] f32

    dim3 grid((OW + TX - 1) / TX, (OH + TY - 1) / TY);  // 64 x 256 tiles
    conv3x3_wmma_kernel<<<grid, 256, 0, stream>>>(X, W, Bv, out);
}